// ShiftNMF_10376640987164
// MI455X (gfx1250) — compile-verified
//
#include <hip/hip_runtime.h>
#include <hip/hip_bf16.h>
#include <stdint.h>

// Problem constants (from reference): N=1024, RANK=16, M=4096.
#define N_DIM  1024
#define R_DIM  16
#define M_DIM  4096
#define M_MASK 4095

// ---------------------------------------------------------------------------
// CDNA5 async global->LDS copy (ASYNCcnt-tracked DMA into LDS, no VGPR data).
// GVS addressing: vdst = LDS byte address VGPR, vaddr = 32-bit byte offset,
// saddr = 64-bit uniform base in SGPR pair.
// ---------------------------------------------------------------------------
__device__ __forceinline__ void async_load_b128(uint32_t lds_byte_addr,
                                                uint32_t glob_byte_off,
                                                const void* glob_base) {
  asm volatile("global_load_async_to_lds_b128 %0, %1, %2"
               :
               : "v"(lds_byte_addr), "v"(glob_byte_off), "s"(glob_base)
               : "memory");
}

__device__ __forceinline__ void wait_async0() {
  asm volatile("s_wait_asynccnt 0" ::: "memory");
}

// ---------------------------------------------------------------------------
// Kernel 1: Hs[r,:] = softmax(H[r,:]) for r in [0,16).  One block per row,
// 256 threads (8 wave32), 16 elements per thread held in registers.
// ---------------------------------------------------------------------------
__global__ __launch_bounds__(256) void softmax_rows_kernel(
    const float* __restrict__ H, float* __restrict__ Hs) {
  const int r   = blockIdx.x;
  const int tid = threadIdx.x;
  const float* __restrict__ row = H + (size_t)r * M_DIM;
  float* __restrict__ orow      = Hs + (size_t)r * M_DIM;

  float v[16];
  float mx = -3.402823466e38f;
#pragma unroll
  for (int j = 0; j < 16; ++j) {
    v[j] = row[tid + (j << 8)];
    mx = fmaxf(mx, v[j]);
  }
  // wave32 reduction
#pragma unroll
  for (int off = 16; off > 0; off >>= 1) mx = fmaxf(mx, __shfl_xor(mx, off));

  __shared__ float sred[8];   // one partial per wave (256/32 = 8)
  __shared__ float sval;
  if ((tid & 31) == 0) sred[tid >> 5] = mx;
  __syncthreads();
  if (tid == 0) {
    float m = sred[0];
#pragma unroll
    for (int i = 1; i < 8; ++i) m = fmaxf(m, sred[i]);
    sval = m;
  }
  __syncthreads();
  mx = sval;

  float sum = 0.f;
#pragma unroll
  for (int j = 0; j < 16; ++j) { v[j] = expf(v[j] - mx); sum += v[j]; }
#pragma unroll
  for (int off = 16; off > 0; off >>= 1) sum += __shfl_xor(sum, off);
  if ((tid & 31) == 0) sred[tid >> 5] = sum;
  __syncthreads();
  if (tid == 0) {
    float s = 0.f;
#pragma unroll
    for (int i = 0; i < 8; ++i) s += sred[i];
    sval = s;
  }
  __syncthreads();
  const float rs = 1.0f / sval;
#pragma unroll
  for (int j = 0; j < 16; ++j) orow[tid + (j << 8)] = v[j] * rs;
}

// ---------------------------------------------------------------------------
// Kernel 2: out[n,m] = sum_r softplus(W[n,r]) * Hs[r, (m - tau[n,r]) & 4095]
// One block per n.  256 threads x 16 m-values each = 4096 outputs.
// Each 16KB Hs row is DMA'd into LDS via global_load_async_to_lds_b128,
// double-buffered: row r+1 streams from L2 while row r is consumed.
// ---------------------------------------------------------------------------
__global__ __launch_bounds__(256) void shiftnmf_kernel(
    const float* __restrict__ W, const int* __restrict__ tau,
    const float* __restrict__ Hs, float* __restrict__ out) {
  const int n   = blockIdx.x;
  const int tid = threadIdx.x;

  __shared__ float s_w[R_DIM];
  __shared__ int   s_tau[R_DIM];
  __shared__ float s_row[2][M_DIM];       // 2 x 16KB double buffer

  if (tid < R_DIM) {
    const float w = W[n * R_DIM + tid];
    // numerically stable softplus: max(w,0) + log1p(exp(-|w|))
    s_w[tid]   = fmaxf(w, 0.f) + log1pf(expf(-fabsf(w)));
    s_tau[tid] = tau[n * R_DIM + tid] & M_MASK;
  }

  // Issue one row's async copy: 256 lanes * 16B * 4 = 16384 bytes.
  auto issue_row = [&](int r, int buf) {
    // flat pointer to LDS: low 32 bits are the LDS byte address
    const uint32_t l = (uint32_t)(uintptr_t)&s_row[buf][0] + (uint32_t)tid * 16u;
    const uint32_t g = (uint32_t)r * (uint32_t)(M_DIM * 4) + (uint32_t)tid * 16u;
#pragma unroll
    for (int k = 0; k < 4; ++k)
      async_load_b128(l + (uint32_t)k * 4096u, g + (uint32_t)k * 4096u, Hs);
  };

  issue_row(0, 0);

  float acc[16];
#pragma unroll
  for (int j = 0; j < 16; ++j) acc[j] = 0.f;

  wait_async0();
  __syncthreads();   // row 0 in LDS; s_w/s_tau visible

#pragma unroll 1
  for (int r = 0; r < R_DIM; ++r) {
    const int cur = r & 1;
    if (r + 1 < R_DIM) issue_row(r + 1, cur ^ 1);   // overlap DMA with compute

    const float w = s_w[r];
    const int   s = s_tau[r];
#pragma unroll
    for (int j = 0; j < 16; ++j) {
      const int m = tid + (j << 8);
      // consecutive tid -> consecutive LDS addresses: bank-conflict-free
      acc[j] = fmaf(w, s_row[cur][(m - s) & M_MASK], acc[j]);
    }

    if (r + 1 < R_DIM) { wait_async0(); __syncthreads(); }
  }

  float* __restrict__ o = out + (size_t)n * M_DIM;
#pragma unroll
  for (int j = 0; j < 16; ++j) o[tid + (j << 8)] = acc[j];   // coalesced
}

// ---------------------------------------------------------------------------
extern "C" void kernel_launch(void* const* d_in, const int* in_sizes, int n_in,
                              void* d_out, int out_size, void* d_ws, size_t ws_size,
                              hipStream_t stream) {
  (void)in_sizes; (void)n_in; (void)out_size; (void)ws_size;
  const float* W   = (const float*)d_in[0];   // (N, R) f32
  const float* H   = (const float*)d_in[1];   // (R, M) f32
  const int*   tau = (const int*)d_in[2];     // (N, R) i32
  float*       out = (float*)d_out;           // (N, M) f32
  float*       Hs  = (float*)d_ws;            // (R, M) f32 scratch = 256 KB

  softmax_rows_kernel<<<R_DIM, 256, 0, stream>>>(H, Hs);
  shiftnmf_kernel<<<N_DIM, 256, 0, stream>>>(W, tau, Hs, out);
}